// CubicFeatureSampling_3487513444832
// MI455X (gfx1250) — compile-verified
//
#include <hip/hip_runtime.h>
#include <hip/hip_bf16.h>
#include <cstdint>
#include <cstddef>

// CubicFeatureSampling for MI455X (gfx1250, wave32).
//  1) transpose_feats: (B,C,V) -> (B,V,C) into d_ws via LDS tiles, staged with
//     CDNA5 async global->LDS DMA (ASYNCcnt + s_wait_asynccnt).
//  2) gather_ws: per-batch launch; one 256-thread block handles 8 points;
//     wave w = corner w; lane l = channels 4l..4l+3 -> one contiguous 512B
//     b128 load per (point, corner), non-temporal b128 store of the 512MiB
//     output. 8-deep unroll lets loads/stores pipeline.
// Memory-bound: ~22us floor from output writes at 23.3 TB/s; the 64MiB
// transposed feature table stays resident in the 192MB L2.

#define GLOBAL_AS __attribute__((address_space(1)))
#define LDS_AS    __attribute__((address_space(3)))

typedef float v4f __attribute__((ext_vector_type(4)));
// Exact element type the async-LDS builtin expects (per hipcc diagnostic):
typedef int v4i __attribute__((__vector_size__(4 * sizeof(int))));

static constexpr int C_CH  = 128;                    // channels (reference)
static constexpr int DGRID = 32;                     // grid dim (reference)
static constexpr int VOX   = DGRID * DGRID * DGRID;  // 32768 voxels
static constexpr int PTS_PER_BLOCK = 8;              // points per workgroup

// ---------------------------------------------------------------------------
// Kernel 1: transpose (B, C, V) -> (B, V, C) through a 32x32 LDS tile.
// Block = 256 threads handles one tile (32 channels x 32 voxels).
// Load phase: 32 rows x 8 chunks of 16B = 256 lane-slots, staged into LDS
// with global_load_async_to_lds_b128.
// ---------------------------------------------------------------------------
__global__ __launch_bounds__(256) void transpose_feats(
    const float* __restrict__ feats, float* __restrict__ ws)
{
  __shared__ float tile[32 * 33];  // pad to 33 floats/row: conflict-free reads
  const int vt = blockIdx.x << 5;  // voxel tile base
  const int ct = blockIdx.y << 5;  // channel tile base
  const int b  = blockIdx.z;
  const int t  = threadIdx.x;

  {
    const int row = t >> 3;   // channel within tile, 0..31
    const int ch  = t & 7;    // 16-byte chunk within row, 0..7
    const float* gsrc =
        feats + (((size_t)b * C_CH + ct + row) * (size_t)VOX + vt + ch * 4);
    float* ldst = &tile[row * 33 + ch * 4];
#if __has_builtin(__builtin_amdgcn_global_load_async_to_lds_b128)
    // CDNA5 async DMA: memory -> LDS, tracked with ASYNCcnt.
    __builtin_amdgcn_global_load_async_to_lds_b128(
        (GLOBAL_AS v4i*)gsrc, (LDS_AS v4i*)ldst, 0, 0);
#else
    *(v4f*)ldst = *(const v4f*)gsrc;
#endif
  }

#if __has_builtin(__builtin_amdgcn_global_load_async_to_lds_b128)
# if __has_builtin(__builtin_amdgcn_s_wait_asynccnt)
  __builtin_amdgcn_s_wait_asynccnt(0);
# else
  asm volatile("s_wait_asynccnt 0x0" ::: "memory");
# endif
#endif
  __syncthreads();

  // Write phase: thread t emits 4 consecutive channels of one voxel.
  // Lanes 0..7 cover channels 0..31 of one voxel -> contiguous 128B stores.
  const int vl = t >> 3;         // voxel within tile, 0..31
  const int c4 = (t & 7) << 2;   // channel quad base, 0..28
  v4f o;
  o.x = tile[(c4 + 0) * 33 + vl];
  o.y = tile[(c4 + 1) * 33 + vl];
  o.z = tile[(c4 + 2) * 33 + vl];
  o.w = tile[(c4 + 3) * 33 + vl];
  *(v4f*)(ws + (((size_t)b * VOX) + vt + vl) * C_CH + ct + c4) = o;
}

// ---------------------------------------------------------------------------
// Corner index + validity (wave-uniform: all 32 lanes of a wave share the
// same point and corner). Returns clamped (always-safe) flat voxel index.
// ---------------------------------------------------------------------------
__device__ __forceinline__ void corner_setup(
    float px, float py, float pz, int dx, int dy, int dz,
    int& v, float& mask)
{
  const float sc = 0.5f * (float)(DGRID - 1);  // 15.5
  int ix = (int)floorf((px + 1.0f) * sc) + dx;
  int iy = (int)floorf((py + 1.0f) * sc) + dy;
  int iz = (int)floorf((pz + 1.0f) * sc) + dz;
  const bool valid = ((unsigned)ix < (unsigned)DGRID) &
                     ((unsigned)iy < (unsigned)DGRID) &
                     ((unsigned)iz < (unsigned)DGRID);
  ix = ix < 0 ? 0 : (ix > DGRID - 1 ? DGRID - 1 : ix);
  iy = iy < 0 ? 0 : (iy > DGRID - 1 ? DGRID - 1 : iy);
  iz = iz < 0 ? 0 : (iz > DGRID - 1 ? DGRID - 1 : iz);
  v = (ix * DGRID + iy) * DGRID + iz;
  mask = valid ? 1.0f : 0.0f;
}

// ---------------------------------------------------------------------------
// Kernel 2: gather from the transposed (V, C) table of one batch.
// Block = 8 points; wave = one corner; lane = 4 channels (b128).
// Unrolled so 8 independent b128 loads pipeline against 8 NT stores.
// ---------------------------------------------------------------------------
__global__ __launch_bounds__(256) void gather_ws(
    const float* __restrict__ pt_b,   // (N,3) of this batch
    const float* __restrict__ ws_b,   // (V,C) of this batch
    float* __restrict__ out_b,        // (N,8,C) of this batch
    int N)
{
  const int base = blockIdx.x * PTS_PER_BLOCK;
  const int o    = threadIdx.x >> 5;  // corner 0..7 (wave id)
  const int lane = threadIdx.x & 31;
  const int dx = (o >> 2) & 1, dy = (o >> 1) & 1, dz = o & 1;

#pragma unroll
  for (int p = 0; p < PTS_PER_BLOCK; ++p) {
    const int n = base + p;
    if (n >= N) break;
    const float px = pt_b[3 * (size_t)n + 0];
    const float py = pt_b[3 * (size_t)n + 1];
    const float pz = pt_b[3 * (size_t)n + 2];
    int v; float m;
    corner_setup(px, py, pz, dx, dy, dz, v, m);

    v4f f = ((const v4f*)(ws_b + (size_t)v * C_CH))[lane];  // 512B/wave
    f.x *= m; f.y *= m; f.z *= m; f.w *= m;
    __builtin_nontemporal_store(
        f, (v4f*)(out_b + ((size_t)n * 8 + o) * C_CH) + lane);
  }
}

// ---------------------------------------------------------------------------
// Fallback: gather straight from (C, V) of one batch if workspace too small.
// Per-lane 4 strided loads (stride = V*4B); L2-resident so still OK.
// ---------------------------------------------------------------------------
__global__ __launch_bounds__(256) void gather_direct(
    const float* __restrict__ pt_b,   // (N,3) of this batch
    const float* __restrict__ feats_b,// (C,V) of this batch
    float* __restrict__ out_b,        // (N,8,C) of this batch
    int N)
{
  const int base = blockIdx.x * PTS_PER_BLOCK;
  const int o    = threadIdx.x >> 5;
  const int lane = threadIdx.x & 31;
  const int dx = (o >> 2) & 1, dy = (o >> 1) & 1, dz = o & 1;
  const int c0 = lane << 2;

#pragma unroll
  for (int p = 0; p < PTS_PER_BLOCK; ++p) {
    const int n = base + p;
    if (n >= N) break;
    const float px = pt_b[3 * (size_t)n + 0];
    const float py = pt_b[3 * (size_t)n + 1];
    const float pz = pt_b[3 * (size_t)n + 2];
    int v; float m;
    corner_setup(px, py, pz, dx, dy, dz, v, m);

    const float* basep = feats_b + v;
    v4f f;
    f.x = basep[(size_t)(c0 + 0) * VOX];
    f.y = basep[(size_t)(c0 + 1) * VOX];
    f.z = basep[(size_t)(c0 + 2) * VOX];
    f.w = basep[(size_t)(c0 + 3) * VOX];
    f.x *= m; f.y *= m; f.z *= m; f.w *= m;
    __builtin_nontemporal_store(
        f, (v4f*)(out_b + ((size_t)n * 8 + o) * C_CH) + lane);
  }
}

extern "C" void kernel_launch(void* const* d_in, const int* in_sizes, int n_in,
                              void* d_out, int out_size, void* d_ws, size_t ws_size,
                              hipStream_t stream)
{
  (void)n_in; (void)out_size;
  const float* ptcloud = (const float*)d_in[0];
  const float* feats   = (const float*)d_in[1];
  float* out = (float*)d_out;

  const int B = in_sizes[1] / (C_CH * VOX);          // 4
  const int total_pts = in_sizes[0] / 3;             // B*N = 131072
  const int N = total_pts / (B > 0 ? B : 1);         // 32768
  const int blocks = (N + PTS_PER_BLOCK - 1) / PTS_PER_BLOCK;  // 4096

  const size_t ws_needed = (size_t)B * VOX * C_CH * sizeof(float);  // 64 MiB
  if (ws_size >= ws_needed) {
    float* wsf = (float*)d_ws;
    dim3 tg(VOX / 32, C_CH / 32, B);                 // (1024, 4, 4)
    transpose_feats<<<tg, 256, 0, stream>>>(feats, wsf);
    for (int b = 0; b < B; ++b) {
      gather_ws<<<blocks, 256, 0, stream>>>(
          ptcloud + (size_t)b * N * 3,
          wsf + (size_t)b * VOX * C_CH,
          out + (size_t)b * N * 8 * C_CH, N);
    }
  } else {
    for (int b = 0; b < B; ++b) {
      gather_direct<<<blocks, 256, 0, stream>>>(
          ptcloud + (size_t)b * N * 3,
          feats + (size_t)b * C_CH * VOX,
          out + (size_t)b * N * 8 * C_CH, N);
    }
  }
}